// HetSANNConv_32238024524292
// MI455X (gfx1250) — compile-verified
//
#include <hip/hip_runtime.h>
#include <hip/hip_bf16.h>

typedef __attribute__((ext_vector_type(2))) float v2f;
typedef __attribute__((ext_vector_type(8))) float v8f;

#define KDIM 128
#define NDIM 128

// Y[n, :] = X[n, :] @ W (+ bias). X:[nrows,128], W:[128(K),128(N)] row-major,
// Y:[npad,128] (rows >= nrows get clamped-garbage, never read).
// Block = 128 threads = 4 waves; each wave owns a 16-row stripe, all 128 cols.
__global__ __launch_bounds__(128) void gemm128_wmma_f32(
    const float* __restrict__ X, const float* __restrict__ W,
    const float* __restrict__ bias, float* __restrict__ Y, int nrows)
{
    const int lane    = threadIdx.x & 31;
    const int wave    = threadIdx.x >> 5;
    const int halfsel = lane >> 4;    // 0: lanes 0-15, 1: lanes 16-31
    const int l16     = lane & 15;

    const long rowbase = ((long)blockIdx.x * 4 + wave) * 16;

    // ---- preload A: row (rowbase + l16); lane half h reads cols 4s + 2h + {0,1}
    long arow = rowbase + l16;
    if (arow > (long)nrows - 1) arow = (long)nrows - 1;
    const float* xrow = X + arow * KDIM + (halfsel << 1);
    v2f a[32];
#pragma unroll
    for (int s = 0; s < 32; ++s)
        a[s] = *(const v2f*)(xrow + 4 * s);

    for (int j = 0; j < 8; ++j) {           // 16-col output tiles
        v8f c = {};
        const float* wb = W + (size_t)(halfsel << 1) * NDIM + j * 16 + l16;
#pragma unroll
        for (int s = 0; s < 32; ++s) {      // K = 128 in steps of 4
            v2f b;
            b.x = wb[(4 * s) * NDIM];
            b.y = wb[(4 * s) * NDIM + NDIM];
            c = __builtin_amdgcn_wmma_f32_16x16x4_f32(false, a[s], false, b,
                                                      (short)0, c, false, false);
        }
        const float bv = bias ? bias[j * 16 + l16] : 0.0f;
#pragma unroll
        for (int i = 0; i < 8; ++i) {
            long r = rowbase + halfsel * 8 + i;  // C layout: M = 8*half + vgpr
            Y[r * NDIM + j * 16 + l16] = c[i] + bv;
        }
    }
}

// For edges with etype==t: emb[dst] += Y[src]. One wave per edge iteration;
// lane l moves 4 consecutive floats (float4 gather, 4 f32 atomics).
__global__ __launch_bounds__(256) void edge_scatter(
    const float* __restrict__ Y, const int* __restrict__ src,
    const int* __restrict__ dst, const int* __restrict__ etype,
    float* __restrict__ emb, int E, int t)
{
    const int lane  = threadIdx.x & 31;
    const int gwave = (int)((blockIdx.x * blockDim.x + threadIdx.x) >> 5);
    const int nwave = (int)((gridDim.x * blockDim.x) >> 5);

    for (int e = gwave; e < E; e += nwave) {
        if (etype[e] != t) continue;               // wave-uniform branch
        const int s = src[e];
        const int d = dst[e];
        const float4 v = *(const float4*)(Y + (size_t)s * NDIM + lane * 4);
        float* p = emb + (size_t)d * NDIM + lane * 4;
        atomicAdd(p + 0, v.x);
        atomicAdd(p + 1, v.y);
        atomicAdd(p + 2, v.z);
        atomicAdd(p + 3, v.w);
    }
}

__global__ __launch_bounds__(256) void elu_kernel(
    const float* __restrict__ emb, float* __restrict__ out, long n)
{
    long i = (long)blockIdx.x * blockDim.x + threadIdx.x;
    long stride = (long)gridDim.x * blockDim.x;
    for (; i < n; i += stride) {
        float v = emb[i];
        out[i] = v > 0.0f ? v : expm1f(v);
    }
}

extern "C" void kernel_launch(void* const* d_in, const int* in_sizes, int n_in,
                              void* d_out, int out_size, void* d_ws, size_t ws_size,
                              hipStream_t stream)
{
    const float* x     = (const float*)d_in[0];
    const int*   src   = (const int*)  d_in[1];
    const int*   dst   = (const int*)  d_in[2];
    const int*   etype = (const int*)  d_in[3];
    const float* W_w   = (const float*)d_in[4];
    // d_in[5] = W_al, d_in[6] = W_ar: dead code in the reference (alpha is deleted)
    const float* W_res = (const float*)d_in[7];
    const float* b_res = (const float*)d_in[8];

    const int N = in_sizes[0] / KDIM;
    const int E = in_sizes[1];
    const int T = in_sizes[4] / (KDIM * NDIM);

    const int Npad    = ((N + 63) / 64) * 64;     // 64 rows per block
    const int gblocks = Npad / 64;

    float* emb = (float*)d_ws;                    // [Npad, 128]
    float* y   = emb + (size_t)Npad * NDIM;       // [Npad, 128], reused per type

    // emb := x @ W_res + b_res  (residual doubles as accumulator init)
    gemm128_wmma_f32<<<gblocks, 128, 0, stream>>>(x, W_res, b_res, emb, N);

    // per-type: y := x @ W_w[t]; then scatter-add y[src] into emb[dst]
    const int eblocks = 1280;                     // 10240 waves scanning E edges
    for (int t = 0; t < T; ++t) {
        gemm128_wmma_f32<<<gblocks, 128, 0, stream>>>(
            x, W_w + (size_t)t * KDIM * NDIM, nullptr, y, N);
        edge_scatter<<<eblocks, 256, 0, stream>>>(y, src, dst, etype, emb, E, t);
    }

    // out = elu(emb)
    const long n = (long)N * NDIM;
    elu_kernel<<<(int)((n + 255) / 256), 256, 0, stream>>>(emb, (float*)d_out, n);
}